// H_3831110828808
// MI455X (gfx1250) — compile-verified
//
#include <hip/hip_runtime.h>

// CDNA5 / gfx1250 lattice hopping kernel:
//   direction > 0: out[b] = roll(U,+1,axis a)^T @ roll(psi,+1,axis a)
//   direction < 0: out[b] = U @ roll(psi,-1,axis a)
//   direction == 0: out = psi
// Memory-bound (~206 MB @ 23.3 TB/s ~= 9 us floor). One wave32 per
// (b,t,x,y) row of 16 z-sites; 4 sites per V_WMMA_F32_16X16X4_F32.

typedef __attribute__((ext_vector_type(2))) float v2f;
typedef __attribute__((ext_vector_type(8))) float v8f;

#define B_ 16
#define T_ 32
#define X_ 16
#define Y_ 16
#define Z_ 16

// strides in float elements
#define F_BS (T_*X_*Y_*Z_*12)   // 1572864
#define F_TS (X_*Y_*Z_*12)      // 49152
#define F_XS (Y_*Z_*12)         // 3072
#define F_YS (Z_*12)            // 192
#define G_AS (T_*X_*Y_*Z_*9)    // 1179648
#define G_TS (X_*Y_*Z_*9)       // 36864
#define G_XS (Y_*Z_*9)          // 2304
#define G_YS (Z_*9)             // 144

__global__ __launch_bounds__(256)
void hop_wmma_f32(const float* __restrict__ field,
                  const float* __restrict__ gauge,
                  const int*   __restrict__ dirp,
                  float*       __restrict__ out)
{
    const int direction = *dirp;          // uniform scalar
    const int tid  = threadIdx.x;
    const int lane = tid & 31;
    const int wv   = tid >> 5;            // wave in block (8 waves)

    if (direction == 0) {
        // identity: float4 grid-stride streaming copy
        const float4* s = (const float4*)field;
        float4*       d = (float4*)out;
        const int total = (B_ * F_BS) / 4;
        for (int i = blockIdx.x * 256 + tid; i < total; i += gridDim.x * 256)
            d[i] = s[i];
        return;
    }

    __shared__ __align__(16) float lds[8 * 528];
    float* ldsF = lds + wv * 528;   // 192 floats: psi row (16 z-sites x 3c x 4s)
    float* ldsG = ldsF + 192;       // 144 floats: U row   (16 z-sites x 3 x 3)
    float* ldsO = ldsG + 144;       // 192 floats: out row

    const int w = blockIdx.x * 8 + wv;    // wave id -> (b,t,x,y)
    const int y = w & 15;
    const int x = (w >> 4) & 15;
    const int t = (w >> 8) & 31;
    const int b = (w >> 13) & 15;

    const int  a     = (direction > 0 ? direction : -direction) - 1; // lattice axis
    const bool pos   = direction > 0;
    const int  delta = pos ? -1 : 1;      // source-coordinate offset (roll semantics)

    int ts = t, xs = x, ys = y, zrotF = 0;
    if      (a == 0) ts = (t + delta + T_) & (T_ - 1);
    else if (a == 1) xs = (x + delta + X_) & (X_ - 1);
    else if (a == 2) ys = (y + delta + Y_) & (Y_ - 1);
    else             zrotF = delta;       // roll along z stays inside this row
    const int tg = pos ? ts : t, xg = pos ? xs : x, yg = pos ? ys : y;
    const int zrotG = (pos && a == 3) ? delta : 0;

    const long fBase = (long)b * F_BS + (long)ts * F_TS + (long)xs * F_XS + (long)ys * F_YS;
    const long gBase = (long)a * G_AS + (long)tg * G_TS + (long)xg * G_XS + (long)yg * G_YS;
    const long oBase = (long)b * F_BS + (long)t  * F_TS + (long)x  * F_XS + (long)y  * F_YS;

    // ---- stage psi row (48 x float4) and U row (36 x float4) into LDS ----
    const float4* fsrc = (const float4*)(field + fBase);   // 768B-aligned row
    const float4* gsrc = (const float4*)(gauge + gBase);   // 576B-aligned row
    for (int k = lane; k < 48; k += 32) ((float4*)ldsF)[k] = fsrc[k];
    for (int k = lane; k < 36; k += 32) ((float4*)ldsG)[k] = gsrc[k];
    asm volatile("s_wait_dscnt 0x0" ::: "memory");  // wave-local LDS RAW fence

    // per-lane constants for operand build (pure functions of lane)
    const int  half = lane >> 4;          // 0: lanes 0-15, 1: lanes 16-31
    const int  m    = lane & 15;          // A row index M
    const int  n    = lane & 15;          // B/D column index N
    const int  nc   = (n < 12) ? n : 11;  // clamp unused columns
    const int  qn   = nc / 3;             // z_local encoded in N
    const int  cpr  = nc - qn * 3;        // color' encoded in N
    const bool ncol = (n < 12);
    const bool hiD  = (qn & 1) != 0;      // D rows for this lane live in VGPRs 4..7
    const bool useA1 = (half == 0);       // A/B VGPR1 is K=3 -> zero for upper half
    const int  k0   = half * 2;           // VGPR0 holds K={0,2}, VGPR1 K={1,3}

    for (int zq = 0; zq < 4; ++zq) {      // 4 WMMAs cover z = 0..15
        // ---- A (16x4 f32): M = (z_local, spin), K = color ----
        const int zA   = zq * 4 + (m >> 2);
        const int sA   = m & 3;
        const int zFA  = (zA + zrotF + Z_) & (Z_ - 1);
        const int aoff = zFA * 12 + sA;                 // psi[c][s] at c*4+s
        const float a0 = ldsF[aoff + half * 8];         // K=0 -> c0, K=2 -> c2
        const float a1 = ldsF[aoff + 4];                // c1 (always in-range)
        v2f A;
        A[0] = a0;
        A[1] = useA1 ? a1 : 0.0f;                       // K=3 -> 0
        // ---- B (4x16 f32): K = color, N = (z_local, color') ----
        const int zB   = zq * 4 + qn;
        const int zGB  = (zB + zrotG + Z_) & (Z_ - 1);
        const int goff = zGB * 9;
        // unconditional in-range loads, then select (avoids exec branches)
        const int i0 = pos ? (k0 * 3 + cpr) : (cpr * 3 + k0);   // U[k0][c'] or U[c'][k0]
        const int i1 = pos ? (3 + cpr)      : (cpr * 3 + 1);    // U[1][c']  or U[c'][1]
        const float b0 = ldsG[goff + i0];
        const float b1 = ldsG[goff + i1];
        v2f Bm;
        Bm[0] = ncol ? b0 : 0.0f;
        Bm[1] = (ncol && useA1) ? b1 : 0.0f;

        v8f C = {};
        // 8-arg form: (neg_a, A, neg_b, B, c_mod, C, reuse_a, reuse_b)
        v8f D = __builtin_amdgcn_wmma_f32_16x16x4_f32(
            false, A, false, Bm, (short)0, C, false, false);

        // ---- scatter block-diagonal D elements to LDS out row ----
        // D VGPR r holds M = r + 8*half, N = lane&15. Useful iff N<12 and
        // the z_local in N matches the z_local in M (block diagonal).
        // For a useful lane, spin s = i lives in VGPR r = 4*(qn&1) + i.
        const bool useful = ncol && ((qn >> 1) == half);
        const int  ob = (zq * 4 + qn) * 12 + cpr * 4;
        const float v0 = hiD ? D[4] : D[0];             // constant-index extracts
        const float v1 = hiD ? D[5] : D[1];
        const float v2 = hiD ? D[6] : D[2];
        const float v3 = hiD ? D[7] : D[3];
        if (useful) {                                   // single exec region per zq
            ldsO[ob + 0] = v0;
            ldsO[ob + 1] = v1;
            ldsO[ob + 2] = v2;
            ldsO[ob + 3] = v3;
        }
    }
    asm volatile("s_wait_dscnt 0x0" ::: "memory");

    // ---- coalesced float4 store of the 192-float output row ----
    float4* odst = (float4*)(out + oBase);
    for (int k = lane; k < 48; k += 32) odst[k] = ((const float4*)ldsO)[k];
}

extern "C" void kernel_launch(void* const* d_in, const int* in_sizes, int n_in,
                              void* d_out, int out_size, void* d_ws, size_t ws_size,
                              hipStream_t stream) {
    (void)in_sizes; (void)n_in; (void)out_size; (void)d_ws; (void)ws_size;
    const float* field = (const float*)d_in[0];
    const float* gauge = (const float*)d_in[1];
    const int*   dir   = (const int*)d_in[2];
    float*       out   = (float*)d_out;
    // one wave per (b,t,x,y): 16*32*16*16 = 131072 waves / 8 waves per block
    hop_wmma_f32<<<16384, 256, 0, stream>>>(field, gauge, dir, out);
}